// DGCNN_81183471829392
// MI455X (gfx1250) — compile-verified
//
#include <hip/hip_runtime.h>

// ============================================================================
// DGCNN forward for MI455X (gfx1250, wave32, WMMA).
//
// Pipeline (all fp32, matching the fp32 reference):
//   per edge layer: sqnorm -> knn (WMMA Gram + top-9) -> [GEMM stats pass]
//                   -> bn finalize -> [GEMM + BN + leaky + max-over-k] -> FEAT
//   fusion: [GEMM stats] -> bn finalize -> [GEMM + BN + leaky + pool(max/mean)]
//   classifier: 3 small FC kernels (BN over batch of 8).
//
// All matrix math uses V_WMMA_F32_16X16X4_F32 (full fp32 fidelity, matrix
// pipe): edge-conv einsum, fusion conv, AND the KNN Gram matrices (the
// largest GEMM in the network, ~70 GFLOP).
// Workspace requirement: ~68 MB (FEAT 64MB + idx/xx/stats/pool buffers).
// ============================================================================

constexpr int BATCH = 8;
constexpr int NPTS  = 4096;
constexpr int KNB   = 9;
constexpr int FEATC = 512;
constexpr float SLOPE = 0.2f;
constexpr float BNEPS = 1e-5f;

typedef float v2f __attribute__((ext_vector_type(2)));
typedef float v8f __attribute__((ext_vector_type(8)));

__device__ __forceinline__ float leakyf(float v) { return v >= 0.f ? v : SLOPE * v; }

// D(16 x 16) = A(16 x 4) * B(4 x 16) + C, fp32 WMMA.
__device__ __forceinline__ v8f wmma4(v2f a, v2f b, v8f c) {
  return __builtin_amdgcn_wmma_f32_16x16x4_f32(false, a, false, b, (short)0, c,
                                               false, false);
}

// Order-preserving float -> uint key (for atomicMax-based float max).
__device__ __forceinline__ unsigned int fkey(float f) {
  unsigned int u = __float_as_uint(f);
  return (u & 0x80000000u) ? ~u : (u | 0x80000000u);
}
__device__ __forceinline__ float funkey(unsigned int k) {
  return __uint_as_float((k & 0x80000000u) ? (k & 0x7FFFFFFFu) : ~k);
}

// ---------------------------------------------------------------------------
// Per-point squared norm: xx[b,n] = sum_c x[b,c,n]^2
// ---------------------------------------------------------------------------
__global__ __launch_bounds__(256) void sqnorm_kernel(
    const float* __restrict__ x, long bstride, int C, float* __restrict__ xx) {
  int t = blockIdx.x * 256 + threadIdx.x;           // over BATCH*NPTS
  int b = t / NPTS, n = t % NPTS;
  const float* xb = x + (long)b * bstride + n;
  float s = 0.f;
  for (int c = 0; c < C; ++c) { float v = xb[(long)c * NPTS]; s += v * v; }
  xx[t] = s;
}

// ---------------------------------------------------------------------------
// KNN via WMMA Gram tiles. WG = 128 thr (4 waves); wave owns 16 query rows,
// A fragments preloaded in registers. Per 64-column tile: stage X cols in
// LDS, 4x (16x16) WMMA score tiles -> LDS, lane-pair scans 32 cols/row with
// sorted top-9 insertion; final 9-way merge of the two half-lists per row.
// ranking key: xx[m] - 2*<x_n, x_m>  (row-constant xx[n] dropped; self skip)
// ---------------------------------------------------------------------------
template <int C>
__global__ __launch_bounds__(128) void knn_kernel(
    const float* __restrict__ x, long bstride,
    const float* __restrict__ xx, int* __restrict__ idxOut) {
  constexpr int CP = (C + 3) & ~3;                  // pad K to multiple of 4
  __shared__ float Ecol[CP * 65];                   // column tile (stride 65)
  __shared__ float sbuf[4 * 16 * 65];               // score tiles per wave
  __shared__ float mdist[128 * KNB];
  __shared__ int   midx[128 * KNB];
  __shared__ float xxs[64];

  int wgr = blockIdx.x % (NPTS / 64);
  int b   = blockIdx.x / (NPTS / 64);
  int tid = threadIdx.x;
  int wv = tid >> 5, L = tid & 31;
  int lhalf = L >> 4, l16 = L & 15;
  int rw   = wgr * 64 + wv * 16;                    // wave's row base
  int rowg = rw + l16;                              // this lane-pair's row
  int cseg = lhalf * 32;
  const float* xb = x + (long)b * bstride;

  // Preload A fragments (16 rows x CP cols), reused for all 64 column tiles.
  v2f av[CP / 4];
#pragma unroll
  for (int ks = 0; ks < CP / 4; ++ks) {
    int c0 = ks * 4 + 2 * lhalf;
    av[ks].x = (c0     < C) ? xb[(long)c0 * NPTS + rw + l16] : 0.f;
    av[ks].y = (c0 + 1 < C) ? xb[(long)(c0 + 1) * NPTS + rw + l16] : 0.f;
  }

  float bd[KNB]; int bi[KNB];
#pragma unroll
  for (int i = 0; i < KNB; ++i) { bd[i] = 3.4e38f; bi[i] = 0; }

  for (int ct = 0; ct < NPTS / 64; ++ct) {
    int colbase = ct * 64;
    __syncthreads();
    for (int e = tid; e < CP * 64; e += 128) {      // stage 64 columns
      int c = e >> 6, mm = e & 63;
      Ecol[c * 65 + mm] = (c < C) ? xb[(long)c * NPTS + colbase + mm] : 0.f;
    }
    if (tid < 64) xxs[tid] = xx[b * NPTS + colbase + tid];
    __syncthreads();
#pragma unroll
    for (int st = 0; st < 4; ++st) {                // 4 x (16x16) score tiles
      v8f acc = {};
#pragma unroll
      for (int ks = 0; ks < CP / 4; ++ks) {
        int krow = ks * 4 + 2 * lhalf;
        v2f bv;
        bv.x = Ecol[krow       * 65 + st * 16 + l16];
        bv.y = Ecol[(krow + 1) * 65 + st * 16 + l16];
        acc = wmma4(av[ks], bv, acc);
      }
#pragma unroll
      for (int r = 0; r < 8; ++r) {
        int rloc = r + 8 * lhalf;
        float sc = xxs[st * 16 + l16] - 2.f * acc[r];
        sbuf[(wv * 16 + rloc) * 65 + st * 16 + l16] = sc;
      }
    }
    __syncthreads();
    for (int mm = 0; mm < 32; ++mm) {               // top-9 insertion scan
      int col = colbase + cseg + mm;
      float sc = sbuf[(wv * 16 + l16) * 65 + cseg + mm];
      if (col == rowg) continue;                    // exclude self
      if (sc < bd[KNB - 1]) {
        int p = KNB - 1;
        while (p > 0 && bd[p - 1] > sc) { bd[p] = bd[p - 1]; bi[p] = bi[p - 1]; --p; }
        bd[p] = sc; bi[p] = col;
      }
    }
  }
  // Merge the two sorted half-lists of each row.
#pragma unroll
  for (int i = 0; i < KNB; ++i) { mdist[tid * KNB + i] = bd[i]; midx[tid * KNB + i] = bi[i]; }
  __syncthreads();
  if (lhalf == 0) {
    int prt = wv * 32 + 16 + l16;                   // partner lane's tid
    int base = (b * NPTS + rowg) * KNB;
    int ia = 0, ib = 0;
    for (int s = 0; s < KNB; ++s) {
      float da = bd[ia];
      float db = mdist[prt * KNB + ib];
      if (da <= db) { idxOut[base + s] = bi[ia]; ++ia; }
      else          { idxOut[base + s] = midx[prt * KNB + ib]; ++ib; }
    }
  }
}

// ---------------------------------------------------------------------------
// Edge conv GEMM: h[o, (n,k)] = W[o, :] . [x_i ; x_j - x_i]
// WG = 128 thr (4 waves), handles (b, 16 n's) -> M = 144 edge columns.
// STATS pass: accumulate per-channel sum / sumsq (atomics).
// Output pass: BN + leaky, max over k, write FEAT slice.
// ---------------------------------------------------------------------------
template <int C, int O, bool STATS>
__global__ __launch_bounds__(128) void edge_kernel(
    const float* __restrict__ x, long bstride, const int* __restrict__ idx,
    const float* __restrict__ W, const float* __restrict__ g,
    const float* __restrict__ beta, const float* __restrict__ mean,
    const float* __restrict__ istd, float* __restrict__ csum,
    float* __restrict__ csq, float* __restrict__ outFeat) {
  constexpr int C2  = 2 * C;
  constexpr int C2P = (C2 + 3) & ~3;                  // pad K to multiple of 4
  constexpr int KS  = (C2P < 128) ? C2P : 128;        // K slab staged in LDS
  constexpr int NSLAB  = C2P / KS;
  constexpr int OC     = (O < 64) ? O : 64;           // channels per chunk
  constexpr int NCHUNK = O / OC;
  constexpr int NWACT  = OC / 16;                     // active waves per chunk

  __shared__ int   sidx[144];
  __shared__ float E[KS * 16];                        // edge tile slab
  __shared__ float hbuf[STATS ? 16 : OC * 144];       // activated h (pass B)

  int nb = blockIdx.x % (NPTS / 16);
  int b  = blockIdx.x / (NPTS / 16);
  int nbase = nb * 16;
  int tid = threadIdx.x;
  int w = tid >> 5, L = tid & 31;
  int lhalf = L >> 4, l16 = L & 15;

  if (tid < 144)
    sidx[tid] = idx[((long)b * NPTS + nbase + tid / KNB) * KNB + tid % KNB];

  const float* xb = x + (long)b * bstride;

  for (int chunk = 0; chunk < NCHUNK; ++chunk) {
    int  otO = chunk * OC + w * 16;                   // wave's o-tile base
    bool act = (w < NWACT);
    float asum[8], asq[8];
#pragma unroll
    for (int r = 0; r < 8; ++r) { asum[r] = 0.f; asq[r] = 0.f; }

    for (int mt = 0; mt < 9; ++mt) {
      v8f acc = {};
      for (int sl = 0; sl < NSLAB; ++sl) {
        __syncthreads();
        for (int e = tid; e < KS * 16; e += 128) {    // stage edge slab
          int cr = e >> 4, mm = e & 15;
          int c2 = sl * KS + cr;
          int m  = mt * 16 + mm;
          int n  = nbase + m / KNB;
          float v = 0.f;
          if (c2 < C) {
            v = xb[(long)c2 * NPTS + n];              // x_i
          } else if (c2 < C2) {
            int c = c2 - C;
            v = xb[(long)c * NPTS + sidx[m]] - xb[(long)c * NPTS + n];  // x_j-x_i
          }
          E[e] = v;
        }
        __syncthreads();
        if (act) {
#pragma unroll
          for (int kk = 0; kk < KS; kk += 4) {
            int c0   = sl * KS + kk + 2 * lhalf;
            int arow = otO + l16;
            v2f a, bv;
            a.x  = (c0     < C2) ? W[(long)arow * C2 + c0]     : 0.f;
            a.y  = (c0 + 1 < C2) ? W[(long)arow * C2 + c0 + 1] : 0.f;
            bv.x = E[(kk + 2 * lhalf)     * 16 + l16];
            bv.y = E[(kk + 2 * lhalf + 1) * 16 + l16];
            acc = wmma4(a, bv, acc);
          }
        }
      }
      if (act) {
        if constexpr (STATS) {
#pragma unroll
          for (int r = 0; r < 8; ++r) { float z = acc[r]; asum[r] += z; asq[r] += z * z; }
        } else {
#pragma unroll
          for (int r = 0; r < 8; ++r) {
            int o = otO + r + 8 * lhalf;
            float z = leakyf((acc[r] - mean[o]) * istd[o] * g[o] + beta[o]);
            hbuf[(w * 16 + r + 8 * lhalf) * 144 + mt * 16 + l16] = z;
          }
        }
      }
    }
    if constexpr (STATS) {
      if (act) {
#pragma unroll
        for (int r = 0; r < 8; ++r) {
          float s = asum[r], q = asq[r];
          for (int d = 1; d < 16; d <<= 1) { s += __shfl_xor(s, d, 16); q += __shfl_xor(q, d, 16); }
          if (l16 == 0) {
            int o = otO + r + 8 * lhalf;
            atomicAdd(&csum[o], s); atomicAdd(&csq[o], q);
          }
        }
      }
    } else {
      __syncthreads();
      for (int t = tid; t < OC * 16; t += 128) {      // max over k=9
        int ol = t / 16, nl = t % 16;
        float mx = -3.4e38f;
#pragma unroll
        for (int k = 0; k < KNB; ++k) {
          float v = hbuf[ol * 144 + nl * KNB + k];
          mx = v > mx ? v : mx;
        }
        outFeat[((long)b * FEATC + chunk * OC + ol) * NPTS + nbase + nl] = mx;
      }
    }
  }
}

// ---------------------------------------------------------------------------
// mean / invstd from accumulated sums
// ---------------------------------------------------------------------------
__global__ void stats_finalize(const float* __restrict__ csum,
                               const float* __restrict__ csq, float cinv, int O,
                               float* __restrict__ mean, float* __restrict__ istd) {
  int o = blockIdx.x * blockDim.x + threadIdx.x;
  if (o < O) {
    float m = csum[o] * cinv;
    float v = csq[o] * cinv - m * m;
    mean[o] = m;
    istd[o] = rsqrtf(v + BNEPS);
  }
}

// ---------------------------------------------------------------------------
// Fusion conv GEMM: z[o, n] = Wf[o, :] . feat[:, n] + bf[o], o in [0,1024)
// WG = 128 thr, (b, 16 n's); E (512x16 feat tile) staged once, reused by
// 64 o-tiles. STATS pass accumulates channel stats; output pass does
// BN+leaky and pools (atomic max-key + atomic sum) over n.
// ---------------------------------------------------------------------------
template <bool STATS>
__global__ __launch_bounds__(128) void fusion_kernel(
    const float* __restrict__ feat, const float* __restrict__ Wf,
    const float* __restrict__ bf, const float* __restrict__ gf,
    const float* __restrict__ betaf, const float* __restrict__ mean,
    const float* __restrict__ istd, float* __restrict__ csum,
    float* __restrict__ csq, unsigned int* __restrict__ Pmax,
    float* __restrict__ Psum) {
  __shared__ float E[512 * 16];                        // 32 KB
  int nb = blockIdx.x % (NPTS / 16);
  int b  = blockIdx.x / (NPTS / 16);
  int nbase = nb * 16;
  int tid = threadIdx.x, w = tid >> 5, L = tid & 31;
  int lhalf = L >> 4, l16 = L & 15;
  const float* fb = feat + (long)b * FEATC * NPTS;
  for (int e = tid; e < 512 * 16; e += 128) {
    int c = e >> 4, mm = e & 15;
    E[e] = fb[(long)c * NPTS + nbase + mm];
  }
  __syncthreads();
  for (int ot = w; ot < 64; ot += 4) {
    v8f acc = {};
    const float* wr = Wf + (long)(ot * 16 + l16) * 512;
#pragma unroll 4
    for (int kk = 0; kk < 512; kk += 4) {
      int c0 = kk + 2 * lhalf;
      v2f a, bv;
      a.x  = wr[c0]; a.y = wr[c0 + 1];
      bv.x = E[c0 * 16 + l16]; bv.y = E[(c0 + 1) * 16 + l16];
      acc = wmma4(a, bv, acc);
    }
#pragma unroll
    for (int r = 0; r < 8; ++r) {
      int o = ot * 16 + r + 8 * lhalf;
      float z = acc[r] + bf[o];
      if constexpr (STATS) {
        float s = z, q = z * z;
        for (int d = 1; d < 16; d <<= 1) { s += __shfl_xor(s, d, 16); q += __shfl_xor(q, d, 16); }
        if (l16 == 0) { atomicAdd(&csum[o], s); atomicAdd(&csq[o], q); }
      } else {
        z = leakyf((z - mean[o]) * istd[o] * gf[o] + betaf[o]);
        float s = z;
        unsigned int mk = fkey(z);
        for (int d = 1; d < 16; d <<= 1) {
          s += __shfl_xor(s, d, 16);
          unsigned int ok = __shfl_xor(mk, d, 16);
          mk = ok > mk ? ok : mk;
        }
        if (l16 == 0) {
          atomicAdd(&Psum[(long)b * 2048 + o], s);     // Psum = P + 1024
          atomicMax(&Pmax[b * 1024 + o], mk);
        }
      }
    }
  }
}

// ---------------------------------------------------------------------------
// Pooling init / finalize. P layout: (b, 2048) = [max(1024) | mean(1024)]
// ---------------------------------------------------------------------------
__global__ void pool_init(unsigned int* __restrict__ Pmax, float* __restrict__ P) {
  int t = blockIdx.x * blockDim.x + threadIdx.x;
  if (t < BATCH * 1024) {
    Pmax[t] = 0u;                                     // below any finite key
    P[(t / 1024) * 2048 + 1024 + (t % 1024)] = 0.f;
  }
}
__global__ void pool_finalize(const unsigned int* __restrict__ Pmax,
                              float* __restrict__ P) {
  int t = blockIdx.x * blockDim.x + threadIdx.x;
  if (t < BATCH * 1024) {
    int b = t / 1024, o = t % 1024;
    P[b * 2048 + o] = funkey(Pmax[t]);
    P[b * 2048 + 1024 + o] *= (1.0f / NPTS);
  }
}

// ---------------------------------------------------------------------------
// Classifier FC: one block per output feature; BN over batch of 8 + leaky.
// ---------------------------------------------------------------------------
template <bool BN>
__global__ __launch_bounds__(64) void fc_kernel(
    const float* __restrict__ in, int IC, const float* __restrict__ Wt,
    const float* __restrict__ bias, const float* __restrict__ g,
    const float* __restrict__ beta, float* __restrict__ out, int OC_) {
  int o = blockIdx.x;
  __shared__ float red[BATCH][64];
  float pr[BATCH];
#pragma unroll
  for (int i = 0; i < BATCH; ++i) pr[i] = 0.f;
  for (int c = threadIdx.x; c < IC; c += 64) {
    float wv = Wt[(long)o * IC + c];
#pragma unroll
    for (int bb = 0; bb < BATCH; ++bb) pr[bb] = fmaf(wv, in[bb * IC + c], pr[bb]);
  }
#pragma unroll
  for (int bb = 0; bb < BATCH; ++bb) red[bb][threadIdx.x] = pr[bb];
  __syncthreads();
  if (threadIdx.x < BATCH) {
    int bb = threadIdx.x;
    float s = 0.f;
    for (int t = 0; t < 64; ++t) s += red[bb][t];
    red[bb][0] = s + bias[o];
  }
  __syncthreads();
  if (threadIdx.x == 0) {
    float z[BATCH], m = 0.f;
#pragma unroll
    for (int bb = 0; bb < BATCH; ++bb) { z[bb] = red[bb][0]; m += z[bb]; }
    if constexpr (BN) {
      m *= (1.f / BATCH);
      float v = 0.f;
#pragma unroll
      for (int bb = 0; bb < BATCH; ++bb) { float d = z[bb] - m; v += d * d; }
      float is = rsqrtf(v * (1.f / BATCH) + BNEPS);
#pragma unroll
      for (int bb = 0; bb < BATCH; ++bb)
        out[bb * OC_ + o] = leakyf((z[bb] - m) * is * g[o] + beta[o]);
    } else {
#pragma unroll
      for (int bb = 0; bb < BATCH; ++bb) out[bb * OC_ + o] = z[bb];
    }
  }
}

// ---------------------------------------------------------------------------
extern "C" void kernel_launch(void* const* d_in, const int* in_sizes, int n_in,
                              void* d_out, int out_size, void* d_ws, size_t ws_size,
                              hipStream_t stream) {
  (void)in_sizes; (void)n_in; (void)out_size; (void)ws_size;

  const float* x   = (const float*)d_in[0];
  // d_in[1] = coords (unused by the reference)
  const float* w1 = (const float*)d_in[2],  *g1 = (const float*)d_in[3],  *b1 = (const float*)d_in[4];
  const float* w2 = (const float*)d_in[5],  *g2 = (const float*)d_in[6],  *b2 = (const float*)d_in[7];
  const float* w3 = (const float*)d_in[8],  *g3 = (const float*)d_in[9],  *b3 = (const float*)d_in[10];
  const float* w4 = (const float*)d_in[11], *g4 = (const float*)d_in[12], *b4 = (const float*)d_in[13];
  const float* wf = (const float*)d_in[14], *bf = (const float*)d_in[15];
  const float* gf = (const float*)d_in[16], *betaf = (const float*)d_in[17];
  const float* wc1 = (const float*)d_in[18], *bc1 = (const float*)d_in[19];
  const float* gc1 = (const float*)d_in[20], *betac1 = (const float*)d_in[21];
  const float* wc2 = (const float*)d_in[22], *bc2 = (const float*)d_in[23];
  const float* gc2 = (const float*)d_in[24], *betac2 = (const float*)d_in[25];
  const float* wc3 = (const float*)d_in[26], *bc3 = (const float*)d_in[27];

  // -------- workspace layout (floats) --------
  float* FEAT = (float*)d_ws;                                  // B*512*N
  int*   IDX  = (int*)(FEAT + (long)BATCH * FEATC * NPTS);     // B*N*9
  float* XX   = (float*)(IDX + (long)BATCH * NPTS * KNB);      // B*N
  float* CSUM = XX + (long)BATCH * NPTS;                       // 1024
  float* CSQ  = CSUM + 1024;                                   // 1024 (contig)
  float* MEAN = CSQ + 1024;                                    // 1024
  float* ISTD = MEAN + 1024;                                   // 1024
  unsigned int* PMAX = (unsigned int*)(ISTD + 1024);           // B*1024
  float* P  = (float*)(PMAX + BATCH * 1024);                   // B*2048
  float* H1 = P + BATCH * 2048;                                // B*512
  float* H2 = H1 + BATCH * 512;                                // B*256

  const int gN256 = BATCH * NPTS / 256;   // 128
  const int gK64  = BATCH * (NPTS / 64);  // 512  (knn WGs)
  const int gT16  = BATCH * (NPTS / 16);  // 2048 (gemm WGs)
  const float cEdge = 1.0f / (float)((long)BATCH * NPTS * KNB);
  const float cFus  = 1.0f / (float)((long)BATCH * NPTS);
  const long  bs0 = 3L * NPTS, bsF = (long)FEATC * NPTS;

  // -------------------- layer 1: x(3) -> FEAT[0:64) --------------------
  sqnorm_kernel<<<gN256, 256, 0, stream>>>(x, bs0, 3, XX);
  knn_kernel<3><<<gK64, 128, 0, stream>>>(x, bs0, XX, IDX);
  hipMemsetAsync(CSUM, 0, 2048 * sizeof(float), stream);
  edge_kernel<3, 64, true ><<<gT16, 128, 0, stream>>>(x, bs0, IDX, w1, g1, b1, MEAN, ISTD, CSUM, CSQ, FEAT);
  stats_finalize<<<1, 256, 0, stream>>>(CSUM, CSQ, cEdge, 64, MEAN, ISTD);
  edge_kernel<3, 64, false><<<gT16, 128, 0, stream>>>(x, bs0, IDX, w1, g1, b1, MEAN, ISTD, CSUM, CSQ, FEAT);

  // -------------------- layer 2: out1(64) -> FEAT[64:128) --------------------
  sqnorm_kernel<<<gN256, 256, 0, stream>>>(FEAT, bsF, 64, XX);
  knn_kernel<64><<<gK64, 128, 0, stream>>>(FEAT, bsF, XX, IDX);
  hipMemsetAsync(CSUM, 0, 2048 * sizeof(float), stream);
  edge_kernel<64, 64, true ><<<gT16, 128, 0, stream>>>(FEAT, bsF, IDX, w2, g2, b2, MEAN, ISTD, CSUM, CSQ, FEAT + 64L * NPTS);
  stats_finalize<<<1, 256, 0, stream>>>(CSUM, CSQ, cEdge, 64, MEAN, ISTD);
  edge_kernel<64, 64, false><<<gT16, 128, 0, stream>>>(FEAT, bsF, IDX, w2, g2, b2, MEAN, ISTD, CSUM, CSQ, FEAT + 64L * NPTS);

  // -------------------- layer 3: out2(64) -> FEAT[128:256) --------------------
  sqnorm_kernel<<<gN256, 256, 0, stream>>>(FEAT + 64L * NPTS, bsF, 64, XX);
  knn_kernel<64><<<gK64, 128, 0, stream>>>(FEAT + 64L * NPTS, bsF, XX, IDX);
  hipMemsetAsync(CSUM, 0, 2048 * sizeof(float), stream);
  edge_kernel<64, 128, true ><<<gT16, 128, 0, stream>>>(FEAT + 64L * NPTS, bsF, IDX, w3, g3, b3, MEAN, ISTD, CSUM, CSQ, FEAT + 128L * NPTS);
  stats_finalize<<<1, 256, 0, stream>>>(CSUM, CSQ, cEdge, 128, MEAN, ISTD);
  edge_kernel<64, 128, false><<<gT16, 128, 0, stream>>>(FEAT + 64L * NPTS, bsF, IDX, w3, g3, b3, MEAN, ISTD, CSUM, CSQ, FEAT + 128L * NPTS);

  // -------------------- layer 4: out3(128) -> FEAT[256:512) --------------------
  sqnorm_kernel<<<gN256, 256, 0, stream>>>(FEAT + 128L * NPTS, bsF, 128, XX);
  knn_kernel<128><<<gK64, 128, 0, stream>>>(FEAT + 128L * NPTS, bsF, XX, IDX);
  hipMemsetAsync(CSUM, 0, 2048 * sizeof(float), stream);
  edge_kernel<128, 256, true ><<<gT16, 128, 0, stream>>>(FEAT + 128L * NPTS, bsF, IDX, w4, g4, b4, MEAN, ISTD, CSUM, CSQ, FEAT + 256L * NPTS);
  stats_finalize<<<1, 256, 0, stream>>>(CSUM, CSQ, cEdge, 256, MEAN, ISTD);
  edge_kernel<128, 256, false><<<gT16, 128, 0, stream>>>(FEAT + 128L * NPTS, bsF, IDX, w4, g4, b4, MEAN, ISTD, CSUM, CSQ, FEAT + 256L * NPTS);

  // -------------------- fusion conv 512 -> 1024 + pooling --------------------
  hipMemsetAsync(CSUM, 0, 2048 * sizeof(float), stream);
  fusion_kernel<true ><<<gT16, 128, 0, stream>>>(FEAT, wf, bf, gf, betaf, MEAN, ISTD, CSUM, CSQ, PMAX, P + 1024);
  stats_finalize<<<4, 256, 0, stream>>>(CSUM, CSQ, cFus, 1024, MEAN, ISTD);
  pool_init<<<(BATCH * 1024 + 255) / 256, 256, 0, stream>>>(PMAX, P);
  fusion_kernel<false><<<gT16, 128, 0, stream>>>(FEAT, wf, bf, gf, betaf, MEAN, ISTD, CSUM, CSQ, PMAX, P + 1024);
  pool_finalize<<<(BATCH * 1024 + 255) / 256, 256, 0, stream>>>(PMAX, P);

  // -------------------- classifier --------------------
  fc_kernel<true ><<<512, 64, 0, stream>>>(P,  2048, wc1, bc1, gc1, betac1, H1, 512);
  fc_kernel<true ><<<256, 64, 0, stream>>>(H1, 512,  wc2, bc2, gc2, betac2, H2, 256);
  fc_kernel<false><<<40,  64, 0, stream>>>(H2, 256,  wc3, bc3, nullptr, nullptr, (float*)d_out, 40);
}